// BasicGNN_27599459844666
// MI455X (gfx1250) — compile-verified
//
#include <hip/hip_runtime.h>
#include <hip/hip_bf16.h>

// ---------------------------------------------------------------------------
// BasicGNN (N=50000, E=800000, D=64, L=3) for MI455X (gfx1250, wave32).
// - GEMMs on V_WMMA_F32_16X16X4_F32 (fp32; workload is HBM-bound, ~35 FLOP/B)
// - A-tiles (edge rows, gathered x[src] rows) staged to LDS with
//   GLOBAL_LOAD_ASYNC_TO_LDS_B128 (ASYNCcnt), read back as ds_load_b64 frags
// - Weights staged into LDS k-pair swizzled so each B fragment is one aligned
//   ds_load_b64 (no register-pairing mov churn)
// ---------------------------------------------------------------------------

#define N_NODES 50000
#define N_EDGES 800000
#define DF      64            // feature dim
#define KSTEPS  (DF / 4)      // 16 WMMA K-steps of K=4
#define NTILES4 (DF / 16)     // 4 column tiles of 16

typedef float v2f __attribute__((ext_vector_type(2)));
typedef float v8f __attribute__((ext_vector_type(8)));

// D = A(16x4) * B(4x16) + C  (fp32 WMMA, wave32)
#define WMMA_F32(a, b, c) \
    __builtin_amdgcn_wmma_f32_16x16x4_f32(false, (a), false, (b), (short)0, (c), false, false)

// ---- gfx1250 async copy: global -> LDS, 16B per lane, tracked by ASYNCcnt ----
__device__ __forceinline__ void async_copy_b128(const float* lds_dst, const float* gsrc) {
    unsigned int l = (unsigned int)(uintptr_t)lds_dst;   // flat addr low 32b == LDS offset
    asm volatile("global_load_async_to_lds_b128 %0, %1, off"
                 :: "v"(l), "v"(gsrc)
                 : "memory");
}
__device__ __forceinline__ void async_wait0() {
    asm volatile("s_wait_asynccnt 0" ::: "memory");
}

// Stage 64x64 weight (row-major [k][n]) into LDS, k-pair swizzled:
// lds[((k>>1)*64 + n)*2 + (k&1)] = W[k][n]  ->  B frag = one ds_load_b64.
__device__ __forceinline__ void stage_weight_sw(float* lds, const float* __restrict__ W,
                                                int tid, int nthreads) {
    for (int i = tid; i < DF * DF; i += nthreads) {
        int k = i >> 6, n = i & 63;
        lds[(((k >> 1) * DF) + n) * 2 + (k & 1)] = W[i];
    }
}

// B fragment (4x16 tile) from swizzled LDS; krow = 4*k + 2*half (even).
__device__ __forceinline__ v2f load_bfrag(const float* lds, int krow, int t, int col) {
    return *(const v2f*)(lds + (((krow >> 1) * DF) + 16 * t + col) * 2);
}

// ---------------------------------------------------------------------------
// Degree kernels
// ---------------------------------------------------------------------------
__global__ __launch_bounds__(256) void deg_count_kernel(const int* __restrict__ dst,
                                                        float* __restrict__ deg) {
    int e = blockIdx.x * 256 + threadIdx.x;
    if (e < N_EDGES) atomicAdd(&deg[dst[e]], 1.0f);
}

__global__ __launch_bounds__(256) void deg_inv_kernel(float* __restrict__ deg) {
    int i = blockIdx.x * 256 + threadIdx.x;
    if (i < N_NODES) {
        float d = deg[i];
        deg[i] = 1.0f / fmaxf(d, 1.0f);
    }
}

// ---------------------------------------------------------------------------
// Conv message + scatter: msg = x[src] @ Wn + ea @ We ; atomic-add into agg[dst]
// One wave per 16-edge tile; 4 waves / block (64KB LDS).
// ---------------------------------------------------------------------------
__global__ __launch_bounds__(128) void conv_edge_kernel(
        const float* __restrict__ x, const float* __restrict__ ea,
        const int* __restrict__ src, const int* __restrict__ dst,
        const float* __restrict__ Wn, const float* __restrict__ We,
        float* __restrict__ agg) {
    __shared__ float lWn[DF * DF];
    __shared__ float lWe[DF * DF];
    __shared__ float lX[4][16 * DF];
    __shared__ float lE[4][16 * DF];
    int tid = threadIdx.x;
    int wave = tid >> 5, lane = tid & 31;
    int tile = blockIdx.x * 4 + wave;
    int e0   = tile * 16;
    int row  = lane & 15, half = lane >> 4, col = lane & 15;

    // async-stage this wave's A tiles (x gather + contiguous ea rows), 16B/lane
    int s = src[e0 + row];
    const float* gx = x  + (size_t)s * DF + half * 32;
    const float* ge = ea + (size_t)(e0 + row) * DF + half * 32;
    float* xt = lX[wave];
    float* et = lE[wave];
    float* xdst = xt + row * DF + half * 32;
    float* edst = et + row * DF + half * 32;
#pragma unroll
    for (int j = 0; j < 8; ++j) {
        async_copy_b128(xdst + j * 4, gx + j * 4);
        async_copy_b128(edst + j * 4, ge + j * 4);
    }

    stage_weight_sw(lWn, Wn, tid, 128);
    stage_weight_sw(lWe, We, tid, 128);
    __syncthreads();     // weights visible to all waves
    async_wait0();       // this wave's tiles landed in LDS

    v8f acc[NTILES4] = {};
#pragma unroll
    for (int k = 0; k < KSTEPS; ++k) {
        int krow = 4 * k + 2 * half;
        v2f ax = *(const v2f*)(xt + row * DF + krow);   // ds_load_b64
        v2f ae = *(const v2f*)(et + row * DF + krow);
#pragma unroll
        for (int t = 0; t < NTILES4; ++t) {
            acc[t] = WMMA_F32(ax, load_bfrag(lWn, krow, t, col), acc[t]);
            acc[t] = WMMA_F32(ae, load_bfrag(lWe, krow, t, col), acc[t]);
        }
    }

    const float* accp = (const float*)acc;   // acc[t] element v at accp[t*8+v]
#pragma unroll
    for (int v = 0; v < 8; ++v) {
        int m = v + 8 * half;                // edge row within tile
        int d = dst[e0 + m];
        float* arow = agg + (size_t)d * DF;
#pragma unroll
        for (int t = 0; t < NTILES4; ++t)
            atomicAdd(arow + 16 * t + col, accp[t * 8 + v]);
    }
}

// ---------------------------------------------------------------------------
// Node GEMM: out = x @ W + bias (+ agg*inv_deg) (+ ReLU)
// One wave per 16-node tile; 8 waves / block (48KB LDS).
// ---------------------------------------------------------------------------
__global__ __launch_bounds__(256) void node_gemm_kernel(
        const float* __restrict__ x, const float* __restrict__ W,
        const float* __restrict__ bias,
        const float* __restrict__ agg, const float* __restrict__ invdeg,
        float* __restrict__ out, int do_relu, int ntiles) {
    __shared__ float lW[DF * DF];
    __shared__ float lX[8][16 * DF];
    int tid = threadIdx.x;
    int wave = tid >> 5, lane = tid & 31;
    int tile = blockIdx.x * 8 + wave;
    int row = lane & 15, half = lane >> 4, col = lane & 15;
    int r0  = tile * 16;
    bool active = (tile < ntiles);

    float* xt = lX[wave];
    if (active) {
        const float* gx = x + (size_t)(r0 + row) * DF + half * 32;
        float* xdst = xt + row * DF + half * 32;
#pragma unroll
        for (int j = 0; j < 8; ++j)
            async_copy_b128(xdst + j * 4, gx + j * 4);
    }

    stage_weight_sw(lW, W, tid, 256);
    __syncthreads();
    if (!active) return;
    async_wait0();

    v8f acc[NTILES4] = {};
#pragma unroll
    for (int k = 0; k < KSTEPS; ++k) {
        int krow = 4 * k + 2 * half;
        v2f a = *(const v2f*)(xt + row * DF + krow);
#pragma unroll
        for (int t = 0; t < NTILES4; ++t)
            acc[t] = WMMA_F32(a, load_bfrag(lW, krow, t, col), acc[t]);
    }

    const float* accp = (const float*)acc;
#pragma unroll
    for (int v = 0; v < 8; ++v) {
        int m = v + 8 * half;
        int node = r0 + m;
        float idg = agg ? invdeg[node] : 0.0f;
#pragma unroll
        for (int t = 0; t < NTILES4; ++t) {
            int nidx = 16 * t + col;
            float val = accp[t * 8 + v] + bias[nidx];
            if (agg) val += agg[(size_t)node * DF + nidx] * idg;
            if (do_relu) val = fmaxf(val, 0.0f);
            out[(size_t)node * DF + nidx] = val;
        }
    }
}

// ---------------------------------------------------------------------------
// Edge MLP: eout = relu(ea@EWe + Ebe + xs[src] + xt[dst]) @ EW1 + Eb1
// One wave per 16-edge tile; 4 waves / block (48KB LDS).
// The ea tile region is reused for the h tile (all reads complete first);
// this converts the WMMA D-layout (lane=col) into A-layout (lane=row).
// ---------------------------------------------------------------------------
__global__ __launch_bounds__(128) void edge_mlp_kernel(
        const float* __restrict__ ea,
        const float* __restrict__ EWe, const float* __restrict__ Ebe,
        const float* __restrict__ xs, const float* __restrict__ xt,
        const int* __restrict__ src, const int* __restrict__ dst,
        const float* __restrict__ EW1, const float* __restrict__ Eb1,
        float* __restrict__ eout) {
    __shared__ float lWe[DF * DF];
    __shared__ float lW1[DF * DF];
    __shared__ float lT[4][16 * DF];     // ea tile, then reused as h tile
    int tid = threadIdx.x;
    int wave = tid >> 5, lane = tid & 31;
    int tile = blockIdx.x * 4 + wave;
    int e0   = tile * 16;
    int row  = lane & 15, half = lane >> 4, col = lane & 15;

    float* et = lT[wave];
    {
        const float* ge = ea + (size_t)(e0 + row) * DF + half * 32;
        float* edst = et + row * DF + half * 32;
#pragma unroll
        for (int j = 0; j < 8; ++j)
            async_copy_b128(edst + j * 4, ge + j * 4);
    }

    stage_weight_sw(lWe, EWe, tid, 128);
    stage_weight_sw(lW1, EW1, tid, 128);
    __syncthreads();
    async_wait0();

    // --- GEMM 1: h = ea @ EWe ---
    v8f acc[NTILES4] = {};
#pragma unroll
    for (int k = 0; k < KSTEPS; ++k) {
        int krow = 4 * k + 2 * half;
        v2f a = *(const v2f*)(et + row * DF + krow);
#pragma unroll
        for (int t = 0; t < NTILES4; ++t)
            acc[t] = WMMA_F32(a, load_bfrag(lWe, krow, t, col), acc[t]);
    }

    // --- epilogue 1: + Ebe + xs[src] + xt[dst], ReLU, stage h-tile in LDS ---
    const float* accp = (const float*)acc;
#pragma unroll
    for (int v = 0; v < 8; ++v) {
        int m  = v + 8 * half;
        int sm = src[e0 + m];
        int dm = dst[e0 + m];
        const float* xsr = xs + (size_t)sm * DF;
        const float* xtr = xt + (size_t)dm * DF;
#pragma unroll
        for (int t = 0; t < NTILES4; ++t) {
            int nidx = 16 * t + col;
            float val = accp[t * 8 + v] + Ebe[nidx] + xsr[nidx] + xtr[nidx];
            et[m * DF + nidx] = fmaxf(val, 0.0f);    // overwrite ea tile
        }
    }

    // --- GEMM 2: eout = h @ EW1 + Eb1 ; A fragments read from LDS ---
    v8f acc2[NTILES4] = {};
#pragma unroll
    for (int k = 0; k < KSTEPS; ++k) {
        int krow = 4 * k + 2 * half;
        v2f a = *(const v2f*)(et + row * DF + krow);
#pragma unroll
        for (int t = 0; t < NTILES4; ++t)
            acc2[t] = WMMA_F32(a, load_bfrag(lW1, krow, t, col), acc2[t]);
    }

    const float* acc2p = (const float*)acc2;
#pragma unroll
    for (int v = 0; v < 8; ++v) {
        int m = v + 8 * half;
        float* orow = eout + (size_t)(e0 + m) * DF;
#pragma unroll
        for (int t = 0; t < NTILES4; ++t) {
            int nidx = 16 * t + col;
            orow[nidx] = acc2p[t * 8 + v] + Eb1[nidx];
        }
    }
}

// ---------------------------------------------------------------------------
// Host launcher
// ---------------------------------------------------------------------------
extern "C" void kernel_launch(void* const* d_in, const int* in_sizes, int n_in,
                              void* d_out, int out_size, void* d_ws, size_t ws_size,
                              hipStream_t stream) {
    const float* x   = (const float*)d_in[0];
    const int*   ei  = (const int*)  d_in[1];
    const float* ea  = (const float*)d_in[2];
    const float* Wr  = (const float*)d_in[3];
    const float* Wn  = (const float*)d_in[4];
    const float* We  = (const float*)d_in[5];
    const float* b   = (const float*)d_in[6];
    const float* EWe = (const float*)d_in[7];
    const float* Ebe = (const float*)d_in[8];
    const float* EWs = (const float*)d_in[9];
    const float* Ebs = (const float*)d_in[10];
    const float* EWt = (const float*)d_in[11];
    const float* Ebt = (const float*)d_in[12];
    const float* EW1 = (const float*)d_in[13];
    const float* Eb1 = (const float*)d_in[14];

    const int* src = ei;
    const int* dst = ei + N_EDGES;

    float* outx = (float*)d_out;                     // [N, D]
    float* oute = outx + (size_t)N_NODES * DF;       // [E, D]

    // Workspace (floats): invdeg[N] | bufA[N*D] | bufB[N*D] | bufC[N*D] | bufD[N*D]
    float* ws     = (float*)d_ws;
    float* invdeg = ws;
    float* bufA   = invdeg + N_NODES;                // x after layer 0
    float* bufB   = bufA + (size_t)N_NODES * DF;     // x after layer 1
    float* bufC   = bufB + (size_t)N_NODES * DF;     // agg, then xs
    float* bufD   = bufC + (size_t)N_NODES * DF;     // xt

    const int NODE_TILES  = N_NODES / 16;            // 3125
    const int NODE_BLOCKS = (NODE_TILES + 7) / 8;    // 391
    const int EDGE_BLOCKS = (N_EDGES / 16) / 4;      // 12500

    // --- degree / inverse degree ---
    hipMemsetAsync(invdeg, 0, (size_t)N_NODES * sizeof(float), stream);
    deg_count_kernel<<<N_EDGES / 256, 256, 0, stream>>>(dst, invdeg);
    deg_inv_kernel<<<(N_NODES + 255) / 256, 256, 0, stream>>>(invdeg);

    const float* xin = x;
    const float* ein = ea;

    for (int i = 0; i < 3; ++i) {
        const float* Wri = Wr + (size_t)i * DF * DF;
        const float* Wni = Wn + (size_t)i * DF * DF;
        const float* Wei = We + (size_t)i * DF * DF;
        const float* bi  = b  + (size_t)i * DF;

        float* agg = bufC;
        hipMemsetAsync(agg, 0, (size_t)N_NODES * DF * sizeof(float), stream);

        conv_edge_kernel<<<EDGE_BLOCKS, 128, 0, stream>>>(xin, ein, src, dst, Wni, Wei, agg);

        float* xout = (i == 0) ? bufA : (i == 1) ? bufB : outx;
        node_gemm_kernel<<<NODE_BLOCKS, 256, 0, stream>>>(
            xin, Wri, bi, agg, invdeg, xout, /*relu=*/(i < 2) ? 1 : 0, NODE_TILES);

        if (i < 2) {
            const float* EWei = EWe + (size_t)i * DF * DF;
            const float* Ebei = Ebe + (size_t)i * DF;
            const float* EWsi = EWs + (size_t)i * DF * DF;
            const float* Ebsi = Ebs + (size_t)i * DF;
            const float* EWti = EWt + (size_t)i * DF * DF;
            const float* Ebti = Ebt + (size_t)i * DF;
            const float* EW1i = EW1 + (size_t)i * DF * DF;
            const float* Eb1i = Eb1 + (size_t)i * DF;

            float* xs = bufC;   // agg no longer needed this layer
            float* xt = bufD;
            node_gemm_kernel<<<NODE_BLOCKS, 256, 0, stream>>>(
                xout, EWsi, Ebsi, nullptr, nullptr, xs, 0, NODE_TILES);
            node_gemm_kernel<<<NODE_BLOCKS, 256, 0, stream>>>(
                xout, EWti, Ebti, nullptr, nullptr, xt, 0, NODE_TILES);

            // layer 0: input ea -> d_out edge region; layer 1: in-place
            edge_mlp_kernel<<<EDGE_BLOCKS, 128, 0, stream>>>(
                ein, EWei, Ebei, xs, xt, src, dst, EW1i, Eb1i, oute);
            ein = oute;
        }
        xin = xout;
    }
}